// Quadratic_4458176053288
// MI455X (gfx1250) — compile-verified
//
#include <hip/hip_runtime.h>

typedef __attribute__((ext_vector_type(2))) float v2f;
typedef __attribute__((ext_vector_type(4))) float v4f;
typedef __attribute__((ext_vector_type(8))) float v8f;
typedef __attribute__((ext_vector_type(4))) int   v4i;

#define DDIM 4096
#define BROWS 16384
#define BM 256          // rows per workgroup
#define BN 256          // cols per workgroup (one N panel)
#define KC 16           // k chunk staged in LDS
#define KPAD 20         // padded row stride of x tile (bank-conflict-free, 16B aligned)
#define NPANELS (DDIM / BN)   // 16

#if __has_builtin(__builtin_amdgcn_global_load_async_to_lds_b128) && \
    __has_builtin(__builtin_amdgcn_s_wait_asynccnt)
#define USE_ASYNC_LDS 1
#else
#define USE_ASYNC_LDS 0
#warning "gfx1250 async-to-LDS builtins not available; falling back to sync staging"
#endif

#if USE_ASYNC_LDS
typedef __attribute__((address_space(1))) v4i* gas_ptr;  // global src: int4 AS(1)*
typedef __attribute__((address_space(3))) v4i* las_ptr;  // LDS dst:    int4 AS(3)*
#endif

// Wave tile: 64 rows x 128 cols = 4 M-tiles x 8 N-tiles of 16x16.
// WG: 8 waves arranged 4 (M) x 2 (N).

__global__ __launch_bounds__(256, 1) void quad_pass1(
    const float* __restrict__ x, const float* __restrict__ A,
    const float* __restrict__ bvec, float* __restrict__ ws)
{
  __shared__ __align__(16) float xs[2][BM][KPAD];   // x[row][k] chunk, padded
  __shared__ __align__(16) float as[2][KC][BN];     // A[k][n] chunk
  __shared__ float obuf[BM];

  const int tid  = threadIdx.x;
  const int lane = tid & 31;
  const int w    = tid >> 5;     // wave 0..7
  const int wm   = w & 3;        // M quadrant
  const int wn   = w >> 2;       // N half
  const int lm   = lane & 15;
  const int h    = lane >> 4;    // half-wave selector (K-pair / row-group)

  const int bn = blockIdx.x & (NPANELS - 1);
  const int bm = blockIdx.x >> 4;
  const int rowbase = bm * BM;
  const int colbase = bn * BN;
  const int wrow = wm * 64;
  const int wcol = wn * 128;

  // staging geometry (per thread: 4 float4 of x-tile, 4 float4 of A-tile)
  const int xr  = tid >> 2, xc4 = tid & 3;        // + j*64 rows
  const int ak  = tid >> 6, ac4 = tid & 63;       // + j*4 k-rows

  v8f acc[4][8];
#pragma unroll
  for (int mt = 0; mt < 4; ++mt)
#pragma unroll
    for (int nt = 0; nt < 8; ++nt)
#pragma unroll
      for (int g = 0; g < 8; ++g) acc[mt][nt][g] = 0.f;

  float lin0 = 0.f, lin1 = 0.f, lin2 = 0.f, lin3 = 0.f;
  const bool doLin = (wn == 0) && (colbase == 0);

  // ---- staging helpers ----
  auto stage = [&](int b, int kb) {
#if USE_ASYNC_LDS
#pragma unroll
    for (int j = 0; j < 4; ++j) {
      const int r = xr + j * 64;
      __builtin_amdgcn_global_load_async_to_lds_b128(
          (gas_ptr)&x[(rowbase + r) * DDIM + kb + xc4 * 4],
          (las_ptr)&xs[b][r][xc4 * 4], 0, 0);
    }
#pragma unroll
    for (int j = 0; j < 4; ++j) {
      const int k = ak + j * 4;
      __builtin_amdgcn_global_load_async_to_lds_b128(
          (gas_ptr)&A[(kb + k) * DDIM + colbase + ac4 * 4],
          (las_ptr)&as[b][k][ac4 * 4], 0, 0);
    }
#else
#pragma unroll
    for (int j = 0; j < 4; ++j) {
      const int r = xr + j * 64;
      *(v4f*)&xs[b][r][xc4 * 4] =
          *(const v4f*)&x[(rowbase + r) * DDIM + kb + xc4 * 4];
    }
#pragma unroll
    for (int j = 0; j < 4; ++j) {
      const int k = ak + j * 4;
      *(v4f*)&as[b][k][ac4 * 4] =
          *(const v4f*)&A[(kb + k) * DDIM + colbase + ac4 * 4];
    }
#endif
  };

  // ---- prologue: stage chunk 0 into buffer 0 ----
  stage(0, 0);

  for (int kc = 0; kc < DDIM; kc += KC) {
    const int buf = (kc >> 4) & 1;
#if USE_ASYNC_LDS
    __builtin_amdgcn_s_wait_asynccnt(0);   // our staged chunk has landed in LDS
#endif
    __syncthreads();
    if (kc + KC < DDIM) stage(buf ^ 1, kc + KC);   // overlap with compute below
#pragma unroll
    for (int k4 = 0; k4 < KC; k4 += 4) {
      // A-matrix (x) fragments: lane holds x[M=lm, k4+2h+{0,1}]
      v2f af[4];
#pragma unroll
      for (int mt = 0; mt < 4; ++mt)
        af[mt] = *(const v2f*)&xs[buf][wrow + mt * 16 + lm][k4 + 2 * h];
      // B-matrix (A) fragments: lane holds A[k4+2h+{0,1}, N=lm]
      v2f bf[8];
#pragma unroll
      for (int nt = 0; nt < 8; ++nt) {
        bf[nt].x = as[buf][k4 + 2 * h + 0][wcol + nt * 16 + lm];
        bf[nt].y = as[buf][k4 + 2 * h + 1][wcol + nt * 16 + lm];
      }
#pragma unroll
      for (int mt = 0; mt < 4; ++mt)
#pragma unroll
        for (int nt = 0; nt < 8; ++nt)
          acc[mt][nt] = __builtin_amdgcn_wmma_f32_16x16x4_f32(
              false, af[mt], false, bf[nt], (short)0, acc[mt][nt],
              false, false);
      if (doLin) {                         // fused x@b using resident fragments
        v2f bb = *(const v2f*)&bvec[kc + k4 + 2 * h];
        lin0 += af[0].x * bb.x + af[0].y * bb.y;
        lin1 += af[1].x * bb.x + af[1].y * bb.y;
        lin2 += af[2].x * bb.x + af[2].y * bb.y;
        lin3 += af[3].x * bb.x + af[3].y * bb.y;
      }
    }
  }

  // ---- epilogue: rowsum = sum_nt acc .* x_cols ----
  // C/D layout: VGPR g, lanes 0-15 -> row g (N=lm), lanes 16-31 -> row g+8.
  float rs[4][8];
#pragma unroll
  for (int mt = 0; mt < 4; ++mt)
#pragma unroll
    for (int g = 0; g < 8; ++g) rs[mt][g] = 0.f;

#pragma unroll
  for (int mt = 0; mt < 4; ++mt)
#pragma unroll
    for (int nt = 0; nt < 8; ++nt) {
      const int col = colbase + wcol + nt * 16 + lm;
#pragma unroll
      for (int g = 0; g < 8; ++g) {
        const int row = rowbase + wrow + mt * 16 + g + 8 * h;
        rs[mt][g] += acc[mt][nt][g] * x[row * DDIM + col];
      }
    }

#pragma unroll
  for (int mt = 0; mt < 4; ++mt)
#pragma unroll
    for (int g = 0; g < 8; ++g) {
      float v = rs[mt][g];
      v += __shfl_xor(v, 1, 32);
      v += __shfl_xor(v, 2, 32);
      v += __shfl_xor(v, 4, 32);
      v += __shfl_xor(v, 8, 32);
      rs[mt][g] = v;          // all 16 lanes of half h hold row (mt*16+g+8h) sum
    }

#if USE_ASYNC_LDS
  __builtin_amdgcn_s_wait_asynccnt(0);
#endif
  __syncthreads();
  obuf[tid] = 0.f;
  __syncthreads();
  if (wn == 0) {
#pragma unroll
    for (int mt = 0; mt < 4; ++mt)
#pragma unroll
      for (int g = 0; g < 8; ++g)
        if (lm == g) obuf[wrow + mt * 16 + g + 8 * h] = rs[mt][g];
  }
  __syncthreads();
  if (wn == 1) {
#pragma unroll
    for (int mt = 0; mt < 4; ++mt)
#pragma unroll
      for (int g = 0; g < 8; ++g)
        if (lm == g) obuf[wrow + mt * 16 + g + 8 * h] += rs[mt][g];
  }
  __syncthreads();
  if (doLin) {
    lin0 += __shfl_xor(lin0, 16, 32);
    lin1 += __shfl_xor(lin1, 16, 32);
    lin2 += __shfl_xor(lin2, 16, 32);
    lin3 += __shfl_xor(lin3, 16, 32);
    if (lane < 16) {
      obuf[wrow + 0 * 16 + lm] += lin0;
      obuf[wrow + 1 * 16 + lm] += lin1;
      obuf[wrow + 2 * 16 + lm] += lin2;
      obuf[wrow + 3 * 16 + lm] += lin3;
    }
  }
  __syncthreads();
  ws[bn * BROWS + rowbase + tid] = obuf[tid];
}

__global__ __launch_bounds__(256) void quad_pass2(
    const float* __restrict__ ws, const float* __restrict__ c,
    float* __restrict__ out)
{
  const int i = blockIdx.x * 256 + threadIdx.x;
  float s = c[0];
#pragma unroll
  for (int p = 0; p < NPANELS; ++p) s += ws[p * BROWS + i];
  out[i] = s;
}

extern "C" void kernel_launch(void* const* d_in, const int* in_sizes, int n_in,
                              void* d_out, int out_size, void* d_ws, size_t ws_size,
                              hipStream_t stream) {
  const float* x = (const float*)d_in[0];
  const float* A = (const float*)d_in[1];
  const float* b = (const float*)d_in[2];
  const float* c = (const float*)d_in[3];
  float* out = (float*)d_out;
  float* ws  = (float*)d_ws;   // needs NPANELS * BROWS * 4 = 1 MB

  dim3 g1((BROWS / BM) * NPANELS);   // 64 M-panels x 16 N-panels = 1024 WGs
  quad_pass1<<<g1, 256, 0, stream>>>(x, A, b, ws);
  quad_pass2<<<BROWS / 256, 256, 0, stream>>>(ws, c, out);
}